// StarSolver_5531917877995
// MI455X (gfx1250) — compile-verified
//
#include <hip/hip_runtime.h>
#include <math.h>

// ---------------------------------------------------------------------------
// StarSolver for MI455X (gfx1250, wave32).
// FIR stage is expressed as polyphase 16x16x16 f32 matmuls and lowered to
// chained V_WMMA_F32_16X16X4_F32 (full f32 precision, matching reference).
// B matrices (banded LSF) are prebuilt in LDS so the WMMA loop is pure
// ds_load + v_wmma with no EXEC-divergent operand construction.
// ---------------------------------------------------------------------------

typedef float v2f __attribute__((ext_vector_type(2)));
typedef float v8f __attribute__((ext_vector_type(8)));

#define C_LIGHT 299792458.0f
#define NX_LSF 16
#define WIN 288           // inner window per 256-output tile: [J0-16, J0+272)
#define SPEC_PER_BLK 8    // 8 waves of 32 -> 256 threads, 1 wave per spectrum

// searchsorted(x, q, side='left') - 1, clamped to [0, N-2], where x[i] = w[i]*f.
// Analytic guess from the (linear) grid + load-based fix-up so the result
// matches comparisons on the actual f32 grid values.
__device__ __forceinline__ int find_ind(const float* __restrict__ w, int N,
                                        float f, float q, float w0, float invDw) {
    float t = (q / f - w0) * invDw;
    int m = (int)floorf(t);
    if (m < 0) m = 0;
    if (m > N - 2) m = N - 2;
    int guard = 0;
    while (m > 0 && w[m] * f >= q && guard++ < 128) --m;        // want x[m] < q
    while (m < N - 2 && w[m + 1] * f < q && guard++ < 128) ++m; // advance while x[m+1] < q
    return m;
}

// ---------------------------------------------------------------------------
// Kernel A: build high-res model (star interp * raw, then 16-tap FIR w/ ones
// padding) for a 256-output tile x 8 spectra per block. FIR via WMMA.
//   model[J0+16t+n] = sum_{u in 0..2} sum_r inner[J0+16(t+u-1)+r] * B_u[r][n]
//   B_u[r][n] = lsf[r + 16(u-1) - n + 7] if index in [0,16) else 0
// ---------------------------------------------------------------------------
__global__ __launch_bounds__(256)
void buildModel(const float* __restrict__ starFlux, const float* __restrict__ starVels,
                const float* __restrict__ raw, const float* __restrict__ waveHr,
                const float* __restrict__ lsf, float* __restrict__ modelOut,
                int nxModel, int nSpec, int s0, int ns) {
    __shared__ float ldsInner[SPEC_PER_BLK][WIN + 1];
    __shared__ float ldsB[SPEC_PER_BLK][3 * NX_LSF * NX_LSF];  // 24 KB
    __shared__ float ldsLsf[NX_LSF][SPEC_PER_BLK];
    __shared__ float ldsF[SPEC_PER_BLK];

    const int J0 = blockIdx.x * 256;
    const int sGroup = blockIdx.y * SPEC_PER_BLK;
    const int tid = threadIdx.x;

    const float w0 = waveHr[0];
    const float invDw = (float)(nxModel - 1) / (waveHr[nxModel - 1] - w0);

    // Phase 1: LSF tile + Doppler factors -> LDS
    if (tid < NX_LSF * SPEC_PER_BLK) {
        int k = tid >> 3, wl = tid & 7;
        int sl = sGroup + wl;
        ldsLsf[k][wl] = (sl < ns) ? lsf[(size_t)k * nSpec + (s0 + sl)] : 0.0f;
    }
    if (tid < SPEC_PER_BLK) {
        int sl = sGroup + tid;
        ldsF[tid] = (sl < ns) ? expf(starVels[s0 + sl] / C_LIGHT) : 1.0f;
    }
    __syncthreads();

    // Phase 2a: banded B matrices (no global traffic, reads ldsLsf)
    for (int e = tid; e < SPEC_PER_BLK * 3 * NX_LSF * NX_LSF; e += 256) {
        int wl = e >> 10;                 // / 768? no: 3*16*16 = 768 -> use div
        int rem = e - wl * 768;
        wl = e / 768; rem = e - wl * 768; // (keep exact; compiler folds)
        int u = rem >> 8;                 // 0..2
        int kk = (rem >> 4) & 15;
        int n = rem & 15;
        int idx = kk + 16 * (u - 1) - n + 7;
        float v = (idx >= 0 && idx < NX_LSF) ? ldsLsf[idx][wl] : 0.0f;
        ldsB[wl][rem] = v;
    }

    // Phase 2b: ones-padded inner window: inner[j] = interp(star)(j) * raw[j][s]
    for (int e = tid; e < SPEC_PER_BLK * WIN; e += 256) {
        int wl = e / WIN;
        int pos = e - wl * WIN;
        int sl = sGroup + wl;
        int j = J0 - 16 + pos;
        float val = 1.0f;  // ones padding (also keeps 0*x finite in WMMA)
        if (sl < ns && j >= 0 && j < nxModel) {
            int s = s0 + sl;
            float f = ldsF[wl];            // shifted grid = waveHr * f
            float q = waveHr[j];
            int m = find_ind(waveHr, nxModel, f, q, w0, invDw);
            float x0 = waveHr[m] * f, x1 = waveHr[m + 1] * f;
            float y0 = starFlux[m], y1 = starFlux[m + 1];
            float st = y0 + (y1 - y0) / (x1 - x0) * (q - x0);
            val = st * raw[(size_t)j * nSpec + s];
        }
        ldsInner[wl][pos] = val;
    }
    __syncthreads();

    // Phase 3: one wave per spectrum, 12 chained f32 WMMAs -> 256 outputs
    const int wv = tid >> 5;          // wave index 0..7
    const int sl = sGroup + wv;
    if (sl >= ns) return;             // wave-uniform exit, EXEC stays all-ones

    const int lane = tid & 31;
    const int tRow = lane & 15;       // A row (t); also B/D column (n)
    const int kHi = (lane >> 4) << 1; // 0 (lanes 0-15) or 2 (lanes 16-31)
    const float* inr = &ldsInner[wv][0];
    const float* bw = &ldsB[wv][0];

    v8f acc = {};
    #pragma unroll
    for (int u = 0; u < 3; ++u) {
        const int rowBase = 16 * (tRow + u - 1) + 16;  // window offset of A[t][r=0]
        const int bBase = (u << 8) + tRow;             // B_u row 0, column n
        #pragma unroll
        for (int c = 0; c < 4; ++c) {
            const int k0 = 4 * c + kHi;
            v2f a;  // A 16x4: lanes 0-15 K=k0,k0+1 ; lanes 16-31 K=k0+2,k0+3
            a.x = inr[rowBase + k0];
            a.y = inr[rowBase + k0 + 1];
            v2f b;  // B 4x16: VGPR0 rows K=k0/k0+2, VGPR1 rows K=k0+1/k0+3
            b.x = bw[bBase + (k0 << 4)];
            b.y = bw[bBase + ((k0 + 1) << 4)];
            acc = __builtin_amdgcn_wmma_f32_16x16x4_f32(
                false, a, false, b, (short)0, acc, false, false);
        }
    }

    // D layout: VGPR v -> M = v + (lane>=16 ? 8 : 0), N = lane&15
    const int tD = (lane >> 4) ? 8 : 0;
    float* row = modelOut + (size_t)sl * nxModel;
    #pragma unroll
    for (int v = 0; v < 8; ++v) {
        int j = J0 + 16 * (v + tD) + tRow;
        if (j < nxModel) row[j] = acc[v];
    }
}

// ---------------------------------------------------------------------------
// Kernel B: resample model onto low-res grid, accumulate weighted sq-diffs
// and weights into per-block partials (deterministic, no float atomics).
// ---------------------------------------------------------------------------
__global__ __launch_bounds__(256)
void lrAccum(const float* __restrict__ model, const float* __restrict__ waveHr,
             const float* __restrict__ waveLr, const float* __restrict__ dataFlux,
             const float* __restrict__ wts, float* __restrict__ partials,
             int nxModel, int nxData, int nSpec, int s0, int ns) {
    __shared__ float r0[256], r1[256];
    int idx = blockIdx.x * 256 + threadIdx.x;
    float a0 = 0.0f, a1 = 0.0f;
    if (idx < nxData * ns) {
        int sl = idx % ns;
        int lr = idx / ns;
        int s = s0 + sl;
        float q = waveLr[(size_t)lr * nSpec + s];
        float w0 = waveHr[0];
        float invDw = (float)(nxModel - 1) / (waveHr[nxModel - 1] - w0);
        int m = find_ind(waveHr, nxModel, 1.0f, q, w0, invDw);
        float x0 = waveHr[m], x1 = waveHr[m + 1];
        const float* row = model + (size_t)sl * nxModel;
        float y0 = row[m], y1 = row[m + 1];
        float mv = y0 + (y1 - y0) / (x1 - x0) * (q - x0);
        float d = dataFlux[(size_t)lr * nSpec + s];
        float wt = wts[(size_t)lr * nSpec + s];
        float e = mv - d;
        a0 = e * e * wt;
        a1 = wt;
    }
    r0[threadIdx.x] = a0;
    r1[threadIdx.x] = a1;
    __syncthreads();
    for (int st = 128; st > 0; st >>= 1) {
        if (threadIdx.x < st) {
            r0[threadIdx.x] += r0[threadIdx.x + st];
            r1[threadIdx.x] += r1[threadIdx.x + st];
        }
        __syncthreads();
    }
    if (threadIdx.x == 0) {
        partials[2 * blockIdx.x] = r0[0];
        partials[2 * blockIdx.x + 1] = r1[0];
    }
}

__global__ void zeroAcc(float* acc) {
    if (threadIdx.x < 2) acc[threadIdx.x] = 0.0f;
}

__global__ __launch_bounds__(256)
void reduceAccum(const float* __restrict__ partials, int nb, float* acc) {
    __shared__ float r0[256], r1[256];
    float a0 = 0.0f, a1 = 0.0f;
    for (int i = threadIdx.x; i < nb; i += 256) {
        a0 += partials[2 * i];
        a1 += partials[2 * i + 1];
    }
    r0[threadIdx.x] = a0;
    r1[threadIdx.x] = a1;
    __syncthreads();
    for (int st = 128; st > 0; st >>= 1) {
        if (threadIdx.x < st) {
            r0[threadIdx.x] += r0[threadIdx.x + st];
            r1[threadIdx.x] += r1[threadIdx.x + st];
        }
        __syncthreads();
    }
    if (threadIdx.x == 0) {
        acc[0] += r0[0];
        acc[1] += r1[0];
    }
}

__global__ void finalOut(const float* acc, float* out) {
    out[0] = sqrtf(acc[0] / acc[1]);
}

// ---------------------------------------------------------------------------
// Host launcher. ws layout: [0,8)=accumulators, [256,16K)=block partials,
// [16K, ...)=model scratch (chunked over spectra to fit any ws_size).
// ---------------------------------------------------------------------------
extern "C" void kernel_launch(void* const* d_in, const int* in_sizes, int n_in,
                              void* d_out, int out_size, void* d_ws, size_t ws_size,
                              hipStream_t stream) {
    const float* starFlux = (const float*)d_in[0];
    const float* starVels = (const float*)d_in[1];
    const float* raw      = (const float*)d_in[2];
    const float* waveLr   = (const float*)d_in[3];
    const float* wts      = (const float*)d_in[4];
    const float* dataFlux = (const float*)d_in[5];
    const float* waveHr   = (const float*)d_in[6];
    const float* lsf      = (const float*)d_in[7];

    const int nxModel = in_sizes[0];          // 262144
    const int nSpec   = in_sizes[1];          // 64
    const int nxData  = in_sizes[3] / nSpec;  // 4096

    float* acc      = (float*)d_ws;
    float* partials = (float*)((char*)d_ws + 256);
    const size_t MODEL_OFF = 16384;
    float* modelBuf = (float*)((char*)d_ws + MODEL_OFF);

    size_t modelBytes = (ws_size > MODEL_OFF) ? (ws_size - MODEL_OFF) : 0;
    size_t perSpec = (size_t)nxModel * sizeof(float);
    int chunk = (perSpec > 0) ? (int)(modelBytes / perSpec) : 1;
    if (chunk > nSpec) chunk = nSpec;
    if (chunk < 1) chunk = 1;

    zeroAcc<<<1, 32, 0, stream>>>(acc);

    for (int s0 = 0; s0 < nSpec; s0 += chunk) {
        int ns = nSpec - s0;
        if (ns > chunk) ns = chunk;

        dim3 gA((nxModel + 255) / 256, (ns + SPEC_PER_BLK - 1) / SPEC_PER_BLK);
        buildModel<<<gA, 256, 0, stream>>>(starFlux, starVels, raw, waveHr, lsf,
                                           modelBuf, nxModel, nSpec, s0, ns);

        int total = nxData * ns;
        int nb = (total + 255) / 256;
        lrAccum<<<nb, 256, 0, stream>>>(modelBuf, waveHr, waveLr, dataFlux, wts,
                                        partials, nxModel, nxData, nSpec, s0, ns);
        reduceAccum<<<1, 256, 0, stream>>>(partials, nb, acc);
    }

    finalOut<<<1, 1, 0, stream>>>(acc, (float*)d_out);
}